// update_v2_52596169507101
// MI455X (gfx1250) — compile-verified
//
#include <hip/hip_runtime.h>

typedef __attribute__((ext_vector_type(2))) float v2f;
typedef __attribute__((ext_vector_type(8))) float v8f;

#define B_    16
#define L_    224
#define K1_   8192
#define P_    16
#define NCH   56        // L_/4 : chunks of K=4 for f32 WMMA
#define SPLIT 32        // blocks per batch in pass 1
#define TPW   2         // k-tiles per wave, pass 1 (32*8*2 = 512 tiles/batch)
#define TPW2  4         // n-tiles per wave, pass 2

// workspace layout (floats): [0,256) sum  [256,512) max-bits  [512,768) mid  [768,1024) h4

__global__ void init_stats_kernel(float* ws) {
    int i = threadIdx.x;            // one block of 256
    ws[i] = 0.0f;                   // sums
    ((unsigned*)ws)[256 + i] = 0u;  // max bits (h3 > 0, so 0 acts as -inf)
}

__global__ __launch_bounds__(256) void pass1_kernel(const float* __restrict__ y,
                                                    const float* __restrict__ w3c,
                                                    const float* __restrict__ sigma,
                                                    float* __restrict__ ws) {
    __shared__ float c_lds[L_ * P_];            // 14 KB: w3_c staged once per block
    const int tid = threadIdx.x;
    for (int i = tid; i < L_ * P_; i += 256) c_lds[i] = w3c[i];   // uniform trip count
    __syncthreads();

    const int lane  = tid & 31;
    const int wave  = tid >> 5;
    const int m     = lane & 15;     // column index (A: M=k_local, B/D: N=p)
    const int kpair = lane >> 4;     // which K-pair this lane holds
    const int loff  = kpair << 1;
    const int b     = blockIdx.x / SPLIT;
    const int split = blockIdx.x % SPLIT;

    // B-matrix (w3_c) chunks resident in VGPRs; fold in c_sq partial
    v2f bch[NCH];
    float csq = 0.0f;
#pragma unroll
    for (int ch = 0; ch < NCH; ++ch) {
        const int l = 4 * ch + loff;
        const float b0 = c_lds[l * P_ + m];
        const float b1 = c_lds[(l + 1) * P_ + m];
        bch[ch][0] = b0; bch[ch][1] = b1;
        csq += b0 * b0 + b1 * b1;
    }
    csq += __shfl_xor(csq, 16, 32);            // full c_sq[p = m]

    const float sg     = sigma[0];
    const float inv2s2 = 0.5f / (sg * sg);

    const float* yb = y + (size_t)b * L_ * K1_;
    float*    sum_ws = ws;
    unsigned* max_ws = (unsigned*)ws + 256;
    float*    mid_ws = ws + 512;

    float s_acc = 0.0f, m_acc = 0.0f;

    for (int t = 0; t < TPW; ++t) {
        const int tile  = (split * 8 + wave) * TPW + t;
        const int kbase = tile * 16;
        v8f acc = {0.f,0.f,0.f,0.f,0.f,0.f,0.f,0.f};
        float ysq = 0.0f;
#pragma unroll
        for (int ch = 0; ch < NCH; ++ch) {
            const int l = 4 * ch + loff;
            const float a0 = yb[(size_t)l       * K1_ + kbase + m];
            const float a1 = yb[(size_t)(l + 1) * K1_ + kbase + m];
            ysq += a0 * a0 + a1 * a1;
            v2f av; av[0] = a0; av[1] = a1;
            acc = __builtin_amdgcn_wmma_f32_16x16x4_f32(
                false, av, false, bch[ch], (short)0, acc, false, false);
        }
        ysq += __shfl_xor(ysq, 16, 32);        // y_sq[kbase + m] on lane-pair m/m+16
#pragma unroll
        for (int r = 0; r < 8; ++r) {
            const int kl = r + (kpair << 3);   // k_local held by this lane in D row r
            const float ysqk = __shfl(ysq, kl, 32);
            float d = ysqk + csq - 2.0f * acc[r];
            d = fmaxf(d, 0.0f);
            const float h = __expf(-d * inv2s2);
            s_acc += h;
            m_acc = fmaxf(m_acc, h);
            if (kbase + kl == K1_ / 2) mid_ws[b * 16 + m] = h;   // lanes 0-15, r==0 tile
        }
    }
    s_acc += __shfl_xor(s_acc, 16, 32);
    m_acc = fmaxf(m_acc, __shfl_xor(m_acc, 16, 32));
    if (lane < 16) {
        atomicAdd(&sum_ws[b * 16 + m], s_acc);
        atomicMax(&max_ws[b * 16 + m], __float_as_uint(m_acc));
    }
}

__global__ __launch_bounds__(256) void finalize_kernel(const float* __restrict__ caw,
                                                       const float* __restrict__ b3w,
                                                       float* __restrict__ ws) {
    __shared__ float combs[256];
    const int tid = threadIdx.x;       // tid = b*16 + p ; 16-lane groups share a batch
    const int b = tid >> 4, p = tid & 15;
    const float avg = ws[b * 16 + p] * (1.0f / (float)K1_);
    const float mx  = __uint_as_float(((unsigned*)ws)[256 + b * 16 + p]);
    const float mid = ws[512 + b * 16 + p];

    float d0a = caw[p]      * avg;
    float d1a = caw[16 + p] * avg;
    float d0m = caw[p]      * mx;
    float d1m = caw[16 + p] * mx;
#pragma unroll
    for (int off = 8; off >= 1; off >>= 1) {
        d0a += __shfl_xor(d0a, off, 16);
        d1a += __shfl_xor(d1a, off, 16);
        d0m += __shfl_xor(d0m, off, 16);
        d1m += __shfl_xor(d1m, off, 16);
    }
    const float z0 = (d0a >= 0.f ? d0a : 0.01f * d0a) + (d0m >= 0.f ? d0m : 0.01f * d0m);
    const float z1 = (d1a >= 0.f ? d1a : 0.01f * d1a) + (d1m >= 0.f ? d1m : 0.01f * d1m);
    const float zm = fmaxf(z0, z1);
    const float e0 = __expf(z0 - zm), e1 = __expf(z1 - zm);
    const float r0 = e0 / (e0 + e1), r1 = 1.0f - r0;
    combs[tid] = r0 * mid + r1 * avg;    // stacked = [mid, mean]
    __syncthreads();
    float h4 = 0.0f;
#pragma unroll
    for (int q = 0; q < 16; ++q) h4 += b3w[p * 16 + q] * combs[b * 16 + q];
    ws[768 + b * 16 + p] = h4;
}

__global__ __launch_bounds__(256) void pass2_kernel(const float* __restrict__ a,
                                                    const float* __restrict__ d2,
                                                    const float* __restrict__ w3w,
                                                    const float* __restrict__ h4ws,
                                                    float* __restrict__ out) {
    const int tid   = threadIdx.x;
    const int lane  = tid & 31;
    const int wave  = tid >> 5;
    const int m     = lane & 15;
    const int kpair = lane >> 4;
    const int loff  = kpair << 1;

    v2f aw[4];                                 // w3_w as A-matrix chunks (M=p, K=q)
#pragma unroll
    for (int ch = 0; ch < 4; ++ch) {
        aw[ch][0] = w3w[m * 16 + 4 * ch + loff];
        aw[ch][1] = w3w[m * 16 + 4 * ch + loff + 1];
    }

    for (int t = 0; t < TPW2; ++t) {
        const int tile  = (blockIdx.x * 8 + wave) * TPW2 + t;
        const int b     = tile >> 9;           // / 512 n-tiles per batch
        const int nbase = (tile & 511) * 16;
        const float h4v = h4ws[b * 16 + m];
        v8f acc;
#pragma unroll
        for (int r = 0; r < 8; ++r)            // C initialized with broadcast h4[b, p]
            acc[r] = __shfl(h4v, r + (kpair << 3), 32);
        const float* ab = a  + ((size_t)b * 16) * K1_;
        const float* db = d2 + ((size_t)b * 16) * K1_;
#pragma unroll
        for (int ch = 0; ch < 4; ++ch) {
            const int q = 4 * ch + loff;
            const size_t o0 = (size_t)q * K1_ + nbase + m;
            const size_t o1 = o0 + K1_;
            v2f bx;
            bx[0] = ab[o0] - db[o0];
            bx[1] = ab[o1] - db[o1];
            acc = __builtin_amdgcn_wmma_f32_16x16x4_f32(
                false, aw[ch], false, bx, (short)0, acc, false, false);
        }
        float* ob = out + ((size_t)b * 16) * K1_ + nbase + m;
#pragma unroll
        for (int r = 0; r < 8; ++r)
            ob[(size_t)(r + (kpair << 3)) * K1_] = acc[r];
    }
}

extern "C" void kernel_launch(void* const* d_in, const int* in_sizes, int n_in,
                              void* d_out, int out_size, void* d_ws, size_t ws_size,
                              hipStream_t stream) {
    const float* a   = (const float*)d_in[0];
    const float* dd2 = (const float*)d_in[1];
    const float* y   = (const float*)d_in[2];
    const float* w3w = (const float*)d_in[3];
    const float* b3w = (const float*)d_in[4];
    const float* w3c = (const float*)d_in[5];
    const float* sig = (const float*)d_in[6];
    const float* caw = (const float*)d_in[7];
    float* out = (float*)d_out;
    float* ws  = (float*)d_ws;

    init_stats_kernel<<<1, 256, 0, stream>>>(ws);
    pass1_kernel<<<B_ * SPLIT, 256, 0, stream>>>(y, w3c, sig, ws);
    finalize_kernel<<<1, 256, 0, stream>>>(caw, b3w, ws);
    pass2_kernel<<<256, 256, 0, stream>>>(a, dd2, w3w, ws + 768, out);
}